// Model_64914135712405
// MI455X (gfx1250) — compile-verified
//
#include <hip/hip_runtime.h>

// ODE scan over 11 steps, elementwise across N points (N = 2^24 in reference).
// d_in: [0]=sig_real (N f32), [1]=sig_imag (N f32), [2]=om (N f32), [3]=ften (11 f32)
// d_out: [sr (N f32) ; si (N f32)] concatenated.

typedef float f32x4 __attribute__((ext_vector_type(4)));

#define ZNUM  11
#define ZI    0.01f
#define ZF    0.99f
#define DELTC ((ZF - ZI) / (float)ZNUM)        // 0.98/11 (fzp denom and step scale)
#define STEPZ ((ZF - ZI) / (float)(ZNUM - 1))  // linspace spacing 0.098
#define MU    1.0f
#define TMC   ((12.0f - MU * MU) / 4.0f)       // 2.75

#ifndef __has_builtin
#define __has_builtin(x) 0
#endif
#if __has_builtin(__builtin_amdgcn_global_load_async_to_lds_b32)
#define HAVE_ASYNC_LDS 1
#else
#define HAVE_ASYNC_LDS 0
#endif

// ---------------------------------------------------------------------------
// Kernel A (1 wave): stage ften into LDS via gfx1250 async global->LDS DMA,
// then compute per-step constants with DELT pre-folded.
// Layout: C[i*8 + {0..5}] = {dc1, a, dc2, d34, dc5, k1}
//   dc1 = D*(-fp/f)        a   = 1/(TM*(1-z))     dc2 = D*a/(1-z)
//   d34 = 1/f^2 - TM^2(1-z)^2                     dc5 = D*mu^2*z^2/f
//   k1  = 1 + dc1
// ---------------------------------------------------------------------------
__global__ void ode_prep_constants(const float* __restrict__ ften,
                                   float* __restrict__ C) {
    __shared__ float lf[ZNUM + 1];
    int i = (int)threadIdx.x;

#if HAVE_ASYNC_LDS
    if (i < ZNUM) {
        __builtin_amdgcn_global_load_async_to_lds_b32(
            (__attribute__((address_space(1))) int*)(ften + i),
            (__attribute__((address_space(3))) int*)(&lf[i]),
            0, 0);
    }
#if __has_builtin(__builtin_amdgcn_s_wait_asynccnt)
    __builtin_amdgcn_s_wait_asynccnt(0);
#else
    asm volatile("s_wait_asynccnt 0" ::: "memory");
#endif
    __syncthreads();
#else
    if (i < ZNUM) lf[i] = ften[i];
    __syncthreads();
#endif

    if (i >= ZNUM) return;
    float z   = ZI + (float)i * STEPZ;
    float f   = lf[i];
    int   j   = (i < ZNUM - 1) ? i : (ZNUM - 2);   // last step reuses backward diff
    float fp  = (lf[j + 1] - lf[j]) / DELTC;
    float omz = 1.0f - z;
    float a   = 1.0f / (TMC * omz);
    float dc1 = DELTC * (-fp / f);
    float dc2 = DELTC * (a / omz);
    float tmz = TMC * omz;
    float d34 = 1.0f / (f * f) - tmz * tmz;
    float dc5 = DELTC * ((MU * MU) * z * z / f);

    float* o = C + i * 8;
    o[0] = dc1; o[1] = a; o[2] = dc2; o[3] = d34; o[4] = dc5;
    o[5] = 1.0f + dc1; o[6] = 0.0f; o[7] = 0.0f;
}

// ---------------------------------------------------------------------------
// One recurrence step, DELT folded into constants. 12 VALU ops.
//   sr' = (dc1 + Dw2*si)*(sr + a) + (sr - dc2)
//   si' = k1*si + Dw*(d34 + si^2 - sr'^2) - a*(Dw2*sr') - dc5*rw
// ---------------------------------------------------------------------------
__device__ __forceinline__ void ode_step(float& sr, float& si,
                                         float rw, float Dw, float Dw2,
                                         float dc1, float a, float dc2,
                                         float d34, float dc5, float k1) {
    float e    = fmaf(Dw2, si, dc1);
    float tt   = sr + a;
    float base = sr - dc2;
    float srn  = fmaf(e, tt, base);

    float acc  = fmaf(Dw, d34, k1 * si);
    float s2   = srn * srn;
    float p    = fmaf(si, si, -s2);
    acc        = fmaf(Dw, p, acc);
    float m    = Dw2 * srn;
    acc        = fmaf(-a, m, acc);
    acc        = fmaf(-dc5, rw, acc);

    sr = srn;
    si = acc;
}

// ---------------------------------------------------------------------------
// Kernel B: streaming scan, 4 elements/thread, NT b128 loads/stores.
// ---------------------------------------------------------------------------
__global__ __launch_bounds__(256) void ode_scan_kernel(
    const float* __restrict__ sig_real, const float* __restrict__ sig_imag,
    const float* __restrict__ om, const float* __restrict__ C,
    float* __restrict__ out_sr, float* __restrict__ out_si, int nvec4) {

    int t = blockIdx.x * blockDim.x + threadIdx.x;
    if (t >= nvec4) return;
    long base = (long)t * 4;

    f32x4 srv = __builtin_nontemporal_load((const f32x4*)(sig_real + base));
    f32x4 siv = __builtin_nontemporal_load((const f32x4*)(sig_imag + base));
    f32x4 wv  = __builtin_nontemporal_load((const f32x4*)(om + base));

    float sr[4] = {srv.x, srv.y, srv.z, srv.w};
    float si[4] = {siv.x, siv.y, siv.z, siv.w};
    float rw[4], Dw[4], Dw2[4];
#pragma unroll
    for (int k = 0; k < 4; ++k) {
        float w = (k == 0) ? wv.x : (k == 1) ? wv.y : (k == 2) ? wv.z : wv.w;
        rw[k]  = __builtin_amdgcn_rcpf(w);   // 1-ulp v_rcp_f32
        Dw[k]  = DELTC * w;
        Dw2[k] = Dw[k] + Dw[k];
    }

#pragma unroll
    for (int s = 0; s < ZNUM; ++s) {
        // wave-uniform -> scalarized to s_load, constants live in SGPRs
        float dc1 = C[s * 8 + 0];
        float a   = C[s * 8 + 1];
        float dc2 = C[s * 8 + 2];
        float d34 = C[s * 8 + 3];
        float dc5 = C[s * 8 + 4];
        float k1  = C[s * 8 + 5];
#pragma unroll
        for (int k = 0; k < 4; ++k)
            ode_step(sr[k], si[k], rw[k], Dw[k], Dw2[k],
                     dc1, a, dc2, d34, dc5, k1);
    }

    f32x4 osr = {sr[0], sr[1], sr[2], sr[3]};
    f32x4 osi = {si[0], si[1], si[2], si[3]};
    __builtin_nontemporal_store(osr, (f32x4*)(out_sr + base));
    __builtin_nontemporal_store(osi, (f32x4*)(out_si + base));
}

// ---------------------------------------------------------------------------
// Tail kernel: scalar handling for n % 4 != 0 (not launched for N = 2^24).
// ---------------------------------------------------------------------------
__global__ void ode_scan_tail(
    const float* __restrict__ sig_real, const float* __restrict__ sig_imag,
    const float* __restrict__ om, const float* __restrict__ C,
    float* __restrict__ out_sr, float* __restrict__ out_si,
    int start, int n) {

    int idx = start + blockIdx.x * blockDim.x + threadIdx.x;
    if (idx >= n) return;
    float sr = sig_real[idx];
    float si = sig_imag[idx];
    float w  = om[idx];
    float rw = __builtin_amdgcn_rcpf(w);
    float Dw = DELTC * w;
    float Dw2 = Dw + Dw;
#pragma unroll
    for (int s = 0; s < ZNUM; ++s)
        ode_step(sr, si, rw, Dw, Dw2,
                 C[s * 8 + 0], C[s * 8 + 1], C[s * 8 + 2],
                 C[s * 8 + 3], C[s * 8 + 4], C[s * 8 + 5]);
    out_sr[idx] = sr;
    out_si[idx] = si;
}

// ---------------------------------------------------------------------------
extern "C" void kernel_launch(void* const* d_in, const int* in_sizes, int n_in,
                              void* d_out, int out_size, void* d_ws, size_t ws_size,
                              hipStream_t stream) {
    const float* sig_real = (const float*)d_in[0];
    const float* sig_imag = (const float*)d_in[1];
    const float* om       = (const float*)d_in[2];
    const float* ften     = (const float*)d_in[3];
    int n = in_sizes[0];

    float* C      = (float*)d_ws;            // ZNUM*8 floats = 352 bytes
    float* out_sr = (float*)d_out;
    float* out_si = out_sr + n;

    ode_prep_constants<<<1, 32, 0, stream>>>(ften, C);

    int nvec4 = n / 4;                       // whole float4 groups
    if (nvec4 > 0) {
        int block = 256;                     // 8 wave32 per block
        int grid  = (nvec4 + block - 1) / block;
        ode_scan_kernel<<<grid, block, 0, stream>>>(sig_real, sig_imag, om, C,
                                                    out_sr, out_si, nvec4);
    }
    int rem_start = nvec4 * 4;
    if (rem_start < n) {
        int rem = n - rem_start;
        ode_scan_tail<<<(rem + 63) / 64, 64, 0, stream>>>(
            sig_real, sig_imag, om, C, out_sr, out_si, rem_start, n);
    }
}